// Attention_29669634081425
// MI455X (gfx1250) — compile-verified
//
#include <hip/hip_runtime.h>

typedef __attribute__((ext_vector_type(2))) float v2f;
typedef __attribute__((ext_vector_type(8))) float v8f;

constexpr int S_LEN = 2048;
constexpr int D_K   = 64;
constexpr int BH    = 32;     // batch*heads = 2*16
constexpr int TQ    = 128;    // q rows per block (8 waves * 16)
constexpr int TKV   = 64;     // kv rows per LDS tile
constexpr int LPAD  = 68;     // LDS row stride (floats), 272B = 16B aligned, bank-skewed
constexpr int NT    = S_LEN / TKV;

#if __has_builtin(__builtin_amdgcn_s_wait_asynccnt)
#define WAIT_ASYNCCNT(n) __builtin_amdgcn_s_wait_asynccnt(n)
#else
#define WAIT_ASYNCCNT(n) asm volatile("s_wait_asynccnt %0" :: "i"(n) : "memory")
#endif

// Flash attention, fp32 end-to-end via V_WMMA_F32_16X16X4_F32.
// Scores computed transposed (S^T = K @ Q^T) so lane == q-row for both the
// score tile and the A-operand of P@V: softmax reductions are register
// reductions + one shfl_xor(16), and the P C->A relayout needs only shuffles.
// K/V tiles are double-buffered in LDS via GLOBAL_LOAD_ASYNC_TO_LDS_B128
// (ASYNCcnt-tracked), so each tile's global latency overlaps a full compute
// iteration. Async loads retire in order per wave: after issuing 8 ops for
// tile t+1, s_wait_asynccnt<=8 guarantees tile t's 8 ops are complete.
__global__ __launch_bounds__(256) void fa_f32_wmma(const float* __restrict__ Qg,
                                                   const float* __restrict__ Kg,
                                                   const float* __restrict__ Vg,
                                                   float* __restrict__ Og) {
  __shared__ float Ks[2][TKV][LPAD];
  __shared__ float Vs[2][TKV][LPAD];

  const int tid   = threadIdx.x;
  const int wave  = tid >> 5;
  const int lane  = tid & 31;
  const int l16   = lane & 15;
  const int lhalf = lane >> 4;           // 0: lanes 0-15, 1: lanes 16-31

  const size_t base = (size_t)blockIdx.y * S_LEN * D_K;
  const float* Q = Qg + base;
  const float* K = Kg + base;
  const float* V = Vg + base;
  float*       O = Og + base;

  const int q0 = blockIdx.x * TQ + wave * 16;

  // Issue one 64x64 K tile + V tile (32KB) as 8 async 16B/lane loads per
  // thread, writing LDS directly (no VGPR staging).  GVS addressing:
  // uniform tile base in SGPR pair, per-lane byte offset in VGPR.
  auto issue_tile = [&](int nbuf, int kv0) {
    const unsigned kbase = (unsigned)(size_t)(&Ks[nbuf][0][0]);
    const unsigned vbase = (unsigned)(size_t)(&Vs[nbuf][0][0]);
    const float* kg = K + (size_t)kv0 * D_K;
    const float* vg = V + (size_t)kv0 * D_K;
    #pragma unroll
    for (int i = 0; i < 4; ++i) {
      const int      idx  = tid + i * 256;              // 1024 float4 / tile
      const unsigned voff = (unsigned)idx * 16u;        // global byte offset
      const unsigned loff =                             // padded LDS offset
          ((unsigned)(idx >> 4) * LPAD + (unsigned)(idx & 15) * 4u) * 4u;
      asm volatile("global_load_async_to_lds_b128 %0, %1, %2"
                   :: "v"(kbase + loff), "v"(voff), "s"(kg) : "memory");
      asm volatile("global_load_async_to_lds_b128 %0, %1, %2"
                   :: "v"(vbase + loff), "v"(voff), "s"(vg) : "memory");
    }
  };

  // Q strip [16 x 64] in B-matrix layout (lane = q col, chunk c holds
  // d = {4c,4c+1} on lanes<16 and {4c+2,4c+3} on lanes>=16).
  // Pre-scale by 8: reference softmax(scores / d_k^-0.5) == softmax(scores*8).
  v2f qB[16];
  {
    const int qrow = q0 + l16;
    #pragma unroll
    for (int c = 0; c < 16; ++c) {
      const int d0 = c * 4 + (lhalf << 1);
      const float2 v = *(const float2*)(Q + (size_t)qrow * D_K + d0);
      qB[c].x = v.x * 8.0f;
      qB[c].y = v.y * 8.0f;
    }
  }

  v8f acc[4] = {};                       // output [q16 x d64], 4 C-tiles
  float m = -1e30f;                      // running row max (per q row)
  float l = 0.0f;                        // running row sum

  issue_tile(0, 0);                      // prologue: tile 0 in flight

  for (int t = 0; t < NT; ++t) {
    const int buf = t & 1;

    __syncthreads();                     // all waves done computing on buf^1
    if (t + 1 < NT) {
      issue_tile(buf ^ 1, (t + 1) * TKV);
      WAIT_ASYNCCNT(8);                  // tile t's 8 ops retired (in-order)
    } else {
      WAIT_ASYNCCNT(0);
    }
    __syncthreads();                     // tile t visible to all waves

    const float (*Kt)[LPAD] = Ks[buf];
    const float (*Vt)[LPAD] = Vs[buf];

    // ---- S^T tiles: st[c] = K_tile_c[16 x 64] @ Q^T[64 x 16] ----
    // C layout: lane -> q = lane%16 ; VGPR j -> kv = c*16 + j + 8*lhalf
    v8f st[4];
    #pragma unroll
    for (int c = 0; c < 4; ++c) {
      v8f s = {};
      const int kvrow = c * 16 + l16;
      #pragma unroll
      for (int ch = 0; ch < 16; ++ch) {
        const int d0 = ch * 4 + (lhalf << 1);
        const float2 kv = *(const float2*)(&Kt[kvrow][d0]);
        v2f a; a.x = kv.x; a.y = kv.y;
        s = __builtin_amdgcn_wmma_f32_16x16x4_f32(false, a, false, qB[ch],
                                                  (short)0, s, false, false);
      }
      st[c] = s;
    }

    // ---- online softmax (each lane owns half the kv range of its q row) ----
    float mt = -1e30f;
    #pragma unroll
    for (int c = 0; c < 4; ++c)
      #pragma unroll
      for (int j = 0; j < 8; ++j) mt = fmaxf(mt, st[c][j]);
    mt = fmaxf(mt, __shfl_xor(mt, 16));
    const float mnew  = fmaxf(m, mt);
    const float alpha = __expf(m - mnew);
    m = mnew;

    float lt = 0.0f;
    #pragma unroll
    for (int c = 0; c < 4; ++c)
      #pragma unroll
      for (int j = 0; j < 8; ++j) {
        const float p = __expf(st[c][j] - mnew);
        st[c][j] = p;
        lt += p;
      }
    lt += __shfl_xor(lt, 16);
    l = l * alpha + lt;

    // rescale accumulator: acc VGPR j corresponds to q row j + 8*lhalf
    {
      const int bsel = lhalf << 3;
      #pragma unroll
      for (int j = 0; j < 8; ++j) {
        const float aj = __shfl(alpha, j + bsel);
        #pragma unroll
        for (int n = 0; n < 4; ++n) acc[n][j] *= aj;
      }
    }

    // ---- P (C layout) -> A layout, then acc += P @ V_tile ----
    #pragma unroll
    for (int c = 0; c < 4; ++c) {
      float x[8];
      #pragma unroll
      for (int j = 0; j < 8; ++j) x[j] = __shfl_xor(st[c][j], 16);
      const bool lo = (lhalf == 0);
      // A chunk cc holds kv {4cc,4cc+1} on lanes<16 and {4cc+2,4cc+3} on lanes>=16
      v2f pa[4];
      pa[0].x = lo ? st[c][0] : x[2];  pa[0].y = lo ? st[c][1] : x[3];
      pa[1].x = lo ? st[c][4] : x[6];  pa[1].y = lo ? st[c][5] : x[7];
      pa[2].x = lo ? x[0] : st[c][2];  pa[2].y = lo ? x[1] : st[c][3];
      pa[3].x = lo ? x[4] : st[c][6];  pa[3].y = lo ? x[5] : st[c][7];

      #pragma unroll
      for (int n = 0; n < 4; ++n) {
        #pragma unroll
        for (int cc = 0; cc < 4; ++cc) {
          const int kvr = c * 16 + cc * 4 + (lhalf << 1);
          v2f b;
          b.x = Vt[kvr][n * 16 + l16];
          b.y = Vt[kvr + 1][n * 16 + l16];
          acc[n] = __builtin_amdgcn_wmma_f32_16x16x4_f32(false, pa[cc], false, b,
                                                         (short)0, acc[n], false, false);
        }
      }
    }
  }

  // ---- normalize by row sums and store ----
  {
    const int bsel = lhalf << 3;
    #pragma unroll
    for (int j = 0; j < 8; ++j) {
      const float inv  = 1.0f / __shfl(l, j + bsel);
      const int   qrow = q0 + j + bsel;
      #pragma unroll
      for (int n = 0; n < 4; ++n)
        O[(size_t)qrow * D_K + n * 16 + l16] = acc[n][j] * inv;
    }
  }
}

extern "C" void kernel_launch(void* const* d_in, const int* in_sizes, int n_in,
                              void* d_out, int out_size, void* d_ws, size_t ws_size,
                              hipStream_t stream) {
  const float* Q = (const float*)d_in[0];
  const float* K = (const float*)d_in[1];
  const float* V = (const float*)d_in[2];
  float*       O = (float*)d_out;
  dim3 grid(S_LEN / TQ, BH);   // 16 q-tiles x 32 (batch*head)
  fa_f32_wmma<<<grid, 256, 0, stream>>>(Q, K, V, O);
}